// VocalTract_15006615732729
// MI455X (gfx1250) — compile-verified
//
#include <hip/hip_runtime.h>

// ---------------- problem constants ----------------
#define TSIZE   44            // tract segments
#define NJ      43            // junctions (SIZE-1)
#define NFRAMES 257
#define FRAME   1024
#define T_CHUNK 512           // steps per chunk (divides FRAME)
#define C_CHUNKS 512          // chunks: C_CHUNKS*T_CHUNK == (NFRAMES-1)*FRAME
#define NTOT    (C_CHUNKS * T_CHUNK)

#define GLOTTAL 0.75f
#define LIPR   (-0.85f)
#define DAMP    0.999f

// ---------------- workspace layout (float offsets) ----------------
#define REFL_OFF  0                        // NFRAMES*NJ = 11051 floats
#define P_OFF     11072                    // start of chunk maps
#define P_DIM     96                       // padded matrix dim (88 state + 1 homog + pad)
#define P_STRIDE  (P_DIM * P_DIM)          // 9216 floats per chunk matrix
#define PREF_OFF  (P_OFF + C_CHUNKS * P_STRIDE)   // boundary states, C_CHUNKS*96 floats

typedef __attribute__((ext_vector_type(2))) float v2f;
typedef __attribute__((ext_vector_type(8))) float v8f;
typedef __attribute__((ext_vector_type(4))) unsigned int u32x4;
typedef __attribute__((ext_vector_type(4))) int i32x4;
typedef __attribute__((ext_vector_type(8))) int i32x8;

// One Kelly-Lochbaum half-step; fully unrolled -> register arrays only.
__device__ __forceinline__ float half_step(float (&R)[TSIZE], float (&L)[TSIZE],
                                           const float (&r)[NJ], float x) {
  float w[NJ];
  float oldR43 = R[TSIZE - 1];
#pragma unroll
  for (int i = 0; i < NJ; ++i) w[i] = r[i] * (R[i] + L[i + 1]);
#pragma unroll
  for (int i = TSIZE - 1; i >= 1; --i) R[i] = (R[i - 1] - w[i - 1]) * DAMP;
  R[0] = fmaf(L[0], GLOTTAL, x) * DAMP;
#pragma unroll
  for (int i = 0; i < NJ; ++i) L[i] = (L[i + 1] + w[i]) * DAMP;
  L[TSIZE - 1] = oldR43 * LIPR * DAMP;
  return R[TSIZE - 1];
}

// ---------------- Phase 0: per-frame reflection coefficients ----------------
__global__ void vt_phase0_refl(const float* __restrict__ diam, float* __restrict__ refl) {
  int idx = blockIdx.x * blockDim.x + threadIdx.x;
  if (idx >= NFRAMES * NJ) return;
  int f = idx / NJ, i = idx - f * NJ;
  float d0 = diam[f * TSIZE + i];
  float d1 = diam[f * TSIZE + i + 1];
  float a0 = d0 * d0, a1 = d1 * d1;
  refl[idx] = (a0 - a1) / (a0 + a1 + 1e-12f);
}

// ---------------- Phase 1: per-chunk affine transfer matrices ----------------
// Block c: 96 threads, thread j builds column j of padded P_c (row-major 96x96).
// Columns 0..87: basis-vector response; column 88: input (inhomogeneous) response;
// columns 89..95 and rows 89..95: zero; P[88][88] = 1 (homogeneous coordinate).
__global__ void vt_phase1_maps(const float* __restrict__ x,
                               const float* __restrict__ refl,
                               float* __restrict__ P) {
  __shared__ float sra[NJ];
  __shared__ float sds[NJ];
  __shared__ float xsh[T_CHUNK];

  const int c = blockIdx.x;
  const int j = threadIdx.x;        // 0..95
  const int f = (c * T_CHUNK) / FRAME;
  const float frac0 = (float)((c * T_CHUNK) % FRAME) * (1.0f / (float)FRAME);

  if (j < NJ) {
    float a = refl[f * NJ + j];
    float b = refl[(f + 1) * NJ + j];
    float d = b - a;
    sra[j] = fmaf(frac0, d, a);               // reflection at chunk start
    sds[j] = d * (1.0f / (float)FRAME);       // per-step slope
  }
  for (int k = j; k < T_CHUNK; k += blockDim.x) xsh[k] = x[c * T_CHUNK + k];
  __syncthreads();

  float R[TSIZE], L[TSIZE];
#pragma unroll
  for (int i = 0; i < TSIZE; ++i) { R[i] = 0.0f; L[i] = 0.0f; }
  if (j < TSIZE) R[j] = 1.0f;
  else if (j < 2 * TSIZE) {
#pragma unroll
    for (int i = 0; i < TSIZE; ++i) if (j - TSIZE == i) L[i] = 1.0f;
  }
  const float xscale = (j == 88) ? 1.0f : 0.0f;

  for (int js = 0; js < T_CHUNK; ++js) {
    float jf = (float)js;
    float r[NJ];
#pragma unroll
    for (int i = 0; i < NJ; ++i) r[i] = fmaf(jf, sds[i], sra[i]);
    float xv = xscale * xsh[js];
    half_step(R, L, r, xv);
    half_step(R, L, r, xv);
  }

  float* Pc = P + (size_t)c * P_STRIDE;
#pragma unroll
  for (int row = 0; row < P_DIM; ++row) {
    float v;
    if (row < TSIZE)           v = R[row];
    else if (row < 2 * TSIZE)  v = L[row - TSIZE];
    else                       v = (row == 88 && j == 88) ? 1.0f : 0.0f;
    Pc[row * P_DIM + j] = v;
  }
}

// -------- TDM: DMA one 96x96 f32 tile (36,864 B) from global into LDS -------
// D# per CDNA5 ISA ch.7.3/8: 2D tile, data_size=4B, tile 96x96, stride 96,
// type=2 ("image"), count=1, no multicast / padding / iteration.
__device__ __forceinline__ void tdm_load_tile(const float* gsrc, unsigned lds_byte_addr) {
  unsigned long long ga = (unsigned long long)(const void*)gsrc;
  u32x4 g0;
  g0[0] = 1u;                                            // count=1 (valid), user mode
  g0[1] = lds_byte_addr;                                 // lds_addr [63:32]
  g0[2] = (unsigned)(ga & 0xFFFFFFFFu);                  // global_addr[31:0]
  g0[3] = (unsigned)((ga >> 32) & 0x01FFFFFFu) | (2u << 30);  // ga[56:32] | type=2
  i32x8 g1;
  g1[0] = (int)(2u << 16);      // workgroup_mask=0, data_size=2 (4 bytes)
  g1[1] = (int)(96u << 16);     // tensor_dim0 = 96   (low16 in bits 63:48)
  g1[2] = (int)(96u << 16);     // tensor_dim0 hi=0, tensor_dim1 = 96 (low16)
  g1[3] = (int)(96u << 16);     // tensor_dim1 hi=0, tile_dim0 = 96
  g1[4] = (int)96;              // tile_dim1 = 96, tile_dim2 = 0
  g1[5] = (int)96;              // tensor_dim0_stride = 96 (low 32)
  g1[6] = 0;                    // stride hi, tensor_dim1_stride lo
  g1[7] = 0;
  i32x4 gz = {0, 0, 0, 0};      // groups 2/3 unused (<=2D tensor)
#if __clang_major__ >= 23
  i32x8 gz8 = {0, 0, 0, 0, 0, 0, 0, 0};
  __builtin_amdgcn_tensor_load_to_lds(g0, g1, gz, gz, gz8, 0);
#else
  __builtin_amdgcn_tensor_load_to_lds(g0, g1, gz, gz, 0);
#endif
}

// ---------------- Phase 2: boundary-state propagation via f32 WMMA ----------
// Single block, 6 waves (192 threads). Wave m owns row-tile m of the 96x96
// matvec s_{c+1} = P_c * s_c, computed as 24 chained V_WMMA_F32_16X16X4_F32.
// P_c tiles are double-buffered in LDS via the Tensor Data Mover so the DMA of
// chunk c+1 overlaps the WMMA chain of chunk c; A-fragments read from LDS.
__global__ void vt_phase2_scan(const float* __restrict__ P,
                               float* __restrict__ prefix) {
  extern __shared__ float smem[];                 // [2*P_STRIDE] tiles + [96] state
  float* Pb0 = smem;
  float* Pb1 = smem + P_STRIDE;
  float* sv  = smem + 2 * P_STRIDE;

  const int tid  = threadIdx.x;
  const int wave = tid >> 5;
  const int lane = tid & 31;
  const int m    = wave;                          // row tile 0..5

  const unsigned lds0 = (unsigned)(size_t)Pb0;
  const unsigned lds1 = (unsigned)(size_t)Pb1;

  if (tid < P_DIM) sv[tid] = (tid == 88) ? 1.0f : 0.0f;   // s_0 = 0 (homog coord = 1)
  if (tid < 32) tdm_load_tile(P, lds0);                    // prime buffer 0 (wave 0)
  __syncthreads();

  for (int c = 0; c < C_CHUNKS; ++c) {
    if (tid < P_DIM) prefix[(size_t)c * P_DIM + tid] = sv[tid];  // state at chunk start

    if (tid < 32) {   // wave 0: keep TDM pipeline one chunk ahead, then drain to c
      if (c + 1 < C_CHUNKS) {
        tdm_load_tile(P + (size_t)(c + 1) * P_STRIDE, ((c + 1) & 1) ? lds1 : lds0);
        __builtin_amdgcn_s_wait_tensorcnt(1);   // in-order TDM: chunk c's tile is in LDS
      } else {
        __builtin_amdgcn_s_wait_tensorcnt(0);
      }
      if (c + 2 < C_CHUNKS)                      // warm L2 two chunks ahead
        __builtin_prefetch(P + (size_t)(c + 2) * P_STRIDE + tid * 288, 0, 1);
    }
    __syncthreads();                             // publish LDS tile to all waves

    const float* Pc = (c & 1) ? Pb1 : Pb0;
    const int rowInTile = lane & 15;
    const int row = m * 16 + rowInTile;
    const int khalf = (lane >> 4) << 1;  // lanes 0-15 -> K{0,1}; lanes 16-31 -> K{2,3}

    v8f acc = {};
    for (int kk = 0; kk < 24; ++kk) {
      const int kbase = kk * 4 + khalf;
      v2f A;
      A.x = Pc[row * P_DIM + kbase];
      A.y = Pc[row * P_DIM + kbase + 1];
      v2f B;
      B.x = 0.0f; B.y = 0.0f;
      if (lane == 0)  { B.x = sv[kk * 4 + 0]; B.y = sv[kk * 4 + 1]; }
      if (lane == 16) { B.x = sv[kk * 4 + 2]; B.y = sv[kk * 4 + 3]; }
      acc = __builtin_amdgcn_wmma_f32_16x16x4_f32(false, A, false, B,
                                                  (short)0, acc, false, false);
    }
    __syncthreads();   // everyone done reading sv and tile buffer c
    // Column N=0 of D: lane 0 holds rows m*16+0..7, lane 16 holds rows m*16+8..15
    if (lane == 0) {
#pragma unroll
      for (int r8 = 0; r8 < 8; ++r8) sv[m * 16 + r8] = acc[r8];
    }
    if (lane == 16) {
#pragma unroll
      for (int r8 = 0; r8 < 8; ++r8) sv[m * 16 + 8 + r8] = acc[r8];
    }
    __syncthreads();   // sv updated for next chunk
  }
}

// ---------------- Phase 3: replay chunks with exact boundary states --------
__global__ void vt_phase3_replay(const float* __restrict__ x,
                                 const float* __restrict__ refl,
                                 const float* __restrict__ prefix,
                                 float* __restrict__ out) {
  __shared__ float sra[64][NJ + 1];
  __shared__ float sds[64][NJ + 1];

  const int tid = threadIdx.x;             // 0..63
  const int c = blockIdx.x * 64 + tid;     // chunk id
  const int f = (c * T_CHUNK) / FRAME;
  const float frac0 = (float)((c * T_CHUNK) % FRAME) * (1.0f / (float)FRAME);

  for (int i = 0; i < NJ; ++i) {
    float a = refl[f * NJ + i];
    float b = refl[(f + 1) * NJ + i];
    float d = b - a;
    sra[tid][i] = fmaf(frac0, d, a);
    sds[tid][i] = d * (1.0f / (float)FRAME);
  }

  float R[TSIZE], L[TSIZE];
  const float* s0 = prefix + (size_t)c * P_DIM;
#pragma unroll
  for (int i = 0; i < TSIZE; ++i) { R[i] = s0[i]; L[i] = s0[TSIZE + i]; }

  const size_t base = (size_t)c * T_CHUNK;
  for (int js = 0; js < T_CHUNK; ++js) {
    float jf = (float)js;
    float r[NJ];
#pragma unroll
    for (int i = 0; i < NJ; ++i) r[i] = fmaf(jf, sds[tid][i], sra[tid][i]);
    float xv = x[base + js];
    float o1 = half_step(R, L, r, xv);
    float o2 = half_step(R, L, r, xv);
    out[base + js] = 0.25f * (o1 + o2);   // == s[1::2]*0.25 of the reference
  }
}

// ---------------- launch ----------------
extern "C" void kernel_launch(void* const* d_in, const int* in_sizes, int n_in,
                              void* d_out, int out_size, void* d_ws, size_t ws_size,
                              hipStream_t stream) {
  (void)in_sizes; (void)n_in; (void)out_size; (void)ws_size;
  const float* input_wave = (const float*)d_in[0];   // (256*1024,)
  const float* diameters  = (const float*)d_in[1];   // (257, 44)
  float* out = (float*)d_out;                        // (256*1024,)

  float* ws     = (float*)d_ws;
  float* refl   = ws + REFL_OFF;
  float* P      = ws + P_OFF;
  float* prefix = ws + PREF_OFF;

  // Phase 0: reflections per frame
  vt_phase0_refl<<<(NFRAMES * NJ + 255) / 256, 256, 0, stream>>>(diameters, refl);
  // Phase 1: 512 chunk transfer matrices (96 columns each)
  vt_phase1_maps<<<C_CHUNKS, 96, 0, stream>>>(input_wave, refl, P);
  // Phase 2: sequential boundary-state scan; TDM double-buffers P_c in LDS,
  // V_WMMA_F32_16X16X4_F32 does the matvecs.
  const size_t shmem = (size_t)(2 * P_STRIDE + P_DIM) * sizeof(float);  // 74,112 B
  vt_phase2_scan<<<1, 192, shmem, stream>>>(P, prefix);
  // Phase 3: parallel replay, emit filtered output
  vt_phase3_replay<<<C_CHUNKS / 64, 64, 0, stream>>>(input_wave, refl, prefix, out);
}